// RPNTrainingProposal_52544629899844
// MI455X (gfx1250) — compile-verified
//
#include <hip/hip_runtime.h>
#include <stdint.h>

typedef float v2f __attribute__((ext_vector_type(2)));
typedef float v8f __attribute__((ext_vector_type(8)));

#define GMAX 64
#define WAVES 8
#define TPB (WAVES * 32)
#define POS_IOU_T 0.7f
#define NEG_IOU_T 0.3f
#define MAX_POS 128
#define TOTAL_SAMPLES 256

// Raw llvm.maxnum/minnum: gfx1250 v_max_num/v_min_num are IEEE-2019 compliant,
// so skip the OCML wrappers (which canonicalize every input -> v_max x,x spam).
#define FMAX(a, b) __builtin_fmaxf((a), (b))
#define FMIN(a, b) __builtin_fminf((a), (b))

// Map float -> unsigned with the same total order (handles negatives: iou can be -1).
__device__ __forceinline__ unsigned fkey(float f) {
  unsigned u = __float_as_uint(f);
  return u ^ (unsigned)(((int)u >> 31) | 0x80000000);
}

__global__ void init_ws(unsigned long long* gtBest, int* cnts) {
  int t = threadIdx.x;
  if (t < GMAX) gtBest[t] = 0ull;
  if (t < 2) cnts[t] = 0;
}

// Fused: IoU (WMMA broadcasts areaA[m] across each 16x16 tile on the XDL pipe),
// row max/argmax, threshold labels, per-gt argmax via packed-u64 LDS/global atomics.
__global__ void __launch_bounds__(TPB)
iou_label_kernel(const float* __restrict__ anchors,
                 const float* __restrict__ gtb,
                 const int* __restrict__ ishape,
                 int N,
                 int* __restrict__ code,                 // per-anchor: ((label+1)<<8)|argmax_g
                 unsigned long long* __restrict__ gtBest) {
  __shared__ __align__(16) float sGt[GMAX * 4];
  __shared__ float sGtArea[GMAX];
  __shared__ unsigned long long sBest[GMAX];
  __shared__ float4 sTile[WAVES][16];     // per-wave anchor tile stage

  const int tid = threadIdx.x;
  const int lane = tid & 31;
  const int wave = tid >> 5;
  const int half = lane >> 4;   // wave32 halves: C/D rows 0-7 vs 8-15
  const int lm = lane & 15;     // N (column) position within tile
  const int half8 = half << 3;

  // --- CDNA5 async copy: gt boxes (64 rows x 16B) global -> LDS ---
  if (tid < GMAX) {
    unsigned ldsOff = (unsigned)(uint64_t)(&sGt[tid * 4]);
    uint64_t ga = (uint64_t)(gtb + tid * 4);
    asm volatile("global_load_async_to_lds_b128 %0, %1, off"
                 :: "v"(ldsOff), "v"(ga) : "memory");
    asm volatile("s_wait_asynccnt 0" ::: "memory");
    sBest[tid] = 0ull;
  }
  __syncthreads();
  if (tid < GMAX) {
    float g0 = sGt[tid * 4 + 0], g1 = sGt[tid * 4 + 1];
    float g2 = sGt[tid * 4 + 2], g3 = sGt[tid * 4 + 3];
    sGtArea[tid] = (g2 - g0) * (g3 - g1);
  }
  __syncthreads();

  const float Hm1 = (float)ishape[0] - 1.0f;
  const float Wm1 = (float)ishape[1] - 1.0f;

  const long long nTiles = ((long long)N + 15) >> 4;
  const long long stride = (long long)gridDim.x * WAVES;

  for (long long t0 = (long long)blockIdx.x * WAVES + wave; t0 < nTiles; t0 += stride) {
    const int base = (int)(t0 << 4);

    if (t0 + stride < nTiles)  // global_prefetch_b8 for next tile
      __builtin_prefetch(anchors + (size_t)((t0 + stride) << 4) * 4, 0, 1);

    __builtin_amdgcn_wave_barrier();   // WAR fence vs previous iteration's reads

    // Lanes 0-15 fetch the 16 anchor rows and stage them in per-wave LDS.
    v2f A; A.y = 0.f;
    if (half == 0) {
      int row = base + lm;
      int rc = (row < N) ? row : (N - 1);
      float4 a = ((const float4*)anchors)[rc];
      sTile[wave][lm] = a;                       // ds_store_b128
      A.x = (a.z - a.x) * (a.w - a.y);           // areaA -> WMMA A (K=0 column)
    } else {
      A.x = 0.f;
    }
    v2f Bv; Bv.x = (half == 0) ? 1.f : 0.f; Bv.y = 0.f;   // B row K=0 = ones

    __builtin_amdgcn_wave_barrier();   // stage visible to whole wave (DS is in-order)

    // One WMMA broadcasts areaA[m] over the tile: D[m][n] = areaA[m] (C = inline 0).
    v8f Cz = {};
    v8f D = __builtin_amdgcn_wmma_f32_16x16x4_f32(
        false, A, false, Bv, (short)0, Cz, false, false);

    // Each lane reads the 8 rows of its half (same-address LDS broadcast reads).
    float rY0[8], rX0[8], rY1[8], rX1[8]; int rOK[8], rIn[8];
#pragma unroll
    for (int r = 0; r < 8; ++r) {
      float4 ar = sTile[wave][r + half8];        // ds_load_b128
      rY0[r] = ar.x; rX0[r] = ar.y; rY1[r] = ar.z; rX1[r] = ar.w;
      rIn[r] = (base + r + half8) < N;
      rOK[r] = rIn[r] && (ar.x >= 0.f) && (ar.y >= 0.f) &&
               (ar.z <= Hm1) && (ar.w <= Wm1);
    }

    float bestIou[8]; int bestG[8];
#pragma unroll
    for (int r = 0; r < 8; ++r) { bestIou[r] = -2.f; bestG[r] = 0; }

#pragma unroll
    for (int t = 0; t < 4; ++t) {           // 4 column tiles cover G=64
      const int n = t * 16 + lm;
      const float4 gv = *(const float4*)(&sGt[n * 4]);   // ds_load_b128
      const float areaBe = sGtArea[n] + 1e-8f;  // fold epsilon with areaB once per tile

      float cbi = -3.f; int cbm = 0;        // column (per-gt) best within this lane
#pragma unroll
      for (int r = 0; r < 8; ++r) {
        float ty = FMAX(rY0[r], gv.x);
        float tx = FMAX(rX0[r], gv.y);
        float by = FMIN(rY1[r], gv.z);
        float bx = FMIN(rX1[r], gv.w);
        float ih = FMAX(by - ty, 0.f);
        float iw = FMAX(bx - tx, 0.f);
        float inter = ih * iw;
        // denom = (areaA[m] - inter) + (areaB[n] + eps); rcp on TRANS pipe.
        float iou = inter * __builtin_amdgcn_rcpf((D[r] - inter) + areaBe);
        iou = rOK[r] ? iou : -1.0f;
        if (iou > bestIou[r]) { bestIou[r] = iou; bestG[r] = n; }     // first-max per lane
        if (rIn[r] && iou > cbi) { cbi = iou; cbm = r + half8; }      // smallest m on tie
      }
      // Pack (orderable key || ~anchorIdx): max => highest IoU, ties => smallest index.
      unsigned long long pack =
          ((unsigned long long)fkey(cbi) << 32) |
          (unsigned long long)(0xFFFFFFFFu - (unsigned)(base + cbm));
      unsigned plo = (unsigned)pack, phi = (unsigned)(pack >> 32);
      unsigned olo = __shfl_xor(plo, 16, 32), ohi = __shfl_xor(phi, 16, 32);
      unsigned long long other = ((unsigned long long)ohi << 32) | olo;
      if (other > pack) pack = other;
      if (half == 0) atomicMax(&sBest[n], pack);    // ds_max_u64
    }

    // Row max/argmax: butterfly over the 16 lanes of each half.
#pragma unroll
    for (int r = 0; r < 8; ++r) {
      float bi = bestIou[r]; int bg = bestG[r];
#pragma unroll
      for (int msk = 1; msk < 16; msk <<= 1) {
        float oi = __shfl_xor(bi, msk, 32);
        int   og = __shfl_xor(bg, msk, 32);
        if (oi > bi || (oi == bi && og < bg)) { bi = oi; bg = og; }
      }
      if (lm == r) {
        int rowIdx = base + r + half8;
        if (rowIdx < N) {
          int lab = -1;
          if (rOK[r]) { if (bi >= POS_IOU_T) lab = 1; else if (bi < NEG_IOU_T) lab = 0; }
          code[rowIdx] = ((lab + 1) << 8) | (bg & 0xFF);
        }
      }
    }
  }

  __syncthreads();
  if (tid < GMAX) {
    unsigned long long p = sBest[tid];
    if (p) atomicMax(&gtBest[tid], p);    // global_atomic_max_u64, order-independent
  }
}

// Best anchor per gt forced positive (sequential => matches .at[].set ordering).
__global__ void force_gt_kernel(const unsigned long long* __restrict__ gtBest,
                                int* __restrict__ code) {
  if (threadIdx.x == 0 && blockIdx.x == 0) {
    for (int g = 0; g < GMAX; ++g) {
      unsigned long long p = gtBest[g];
      if (p) {
        int idx = (int)(0xFFFFFFFFu - (unsigned)(p & 0xFFFFFFFFu));
        code[idx] = (2 << 8) | g;   // label=1, argmax=g
      }
    }
  }
}

// Fixed-size subsampling (atomic ticket selection; shape semantics match reference).
__global__ void sample_kernel(const int* __restrict__ code, int N,
                              int* cnts, int* posIdx, int* negIdx) {
  long long i0 = (long long)blockIdx.x * blockDim.x + threadIdx.x;
  long long st = (long long)gridDim.x * blockDim.x;
  for (long long i = i0; i < N; i += st) {
    int lab = (code[(int)i] >> 8) - 1;
    if (lab == 1) {
      int t = atomicAdd(&cnts[0], 1);
      if (t < MAX_POS) posIdx[t] = (int)i;
    } else if (lab == 0) {
      int t = atomicAdd(&cnts[1], 1);
      if (t < TOTAL_SAMPLES) negIdx[t] = (int)i;
    }
  }
}

// Assemble outputs: idx[256], labels[256], txtytwth[128*4], cur_pos.
__global__ void finalize_kernel(const float* __restrict__ anchors,
                                const float* __restrict__ gtb,
                                const int* __restrict__ code,
                                const int* __restrict__ posIdx,
                                const int* __restrict__ negIdx,
                                const int* __restrict__ cnts,
                                float* __restrict__ out) {
  int s = threadIdx.x;  // 256 threads
  int cp = min(cnts[0], MAX_POS);
  int cn = min(cnts[1], TOTAL_SAMPLES - cp);
  int tai = -1, lab = -1;
  if (s < cp)            { tai = posIdx[s];      lab = 1; }
  else if (s < cp + cn)  { tai = negIdx[s - cp]; lab = 0; }
  out[s] = (float)tai;
  out[TOTAL_SAMPLES + s] = (float)lab;

  if (s < MAX_POS) {
    float t0 = 0.f, t1 = 0.f, t2 = 0.f, t3 = 0.f;
    if (s < cp) {
      int ai = posIdx[s];
      float4 a  = ((const float4*)anchors)[ai];
      int   g   = code[ai] & 0xFF;
      float4 gv = ((const float4*)gtb)[g];
      float ah = a.z - a.x, aw = a.w - a.y;
      float ay = a.x + 0.5f * ah, ax = a.y + 0.5f * aw;
      float gh = gv.z - gv.x, gw = gv.w - gv.y;
      float gy = gv.x + 0.5f * gh, gx = gv.y + 0.5f * gw;
      t0 = (gy - ay) / ah; t1 = (gx - ax) / aw;
      t2 = __logf(gh / ah); t3 = __logf(gw / aw);
    }
    out[2 * TOTAL_SAMPLES + s * 4 + 0] = t0;
    out[2 * TOTAL_SAMPLES + s * 4 + 1] = t1;
    out[2 * TOTAL_SAMPLES + s * 4 + 2] = t2;
    out[2 * TOTAL_SAMPLES + s * 4 + 3] = t3;
  }
  if (s == 0) out[2 * TOTAL_SAMPLES + 4 * MAX_POS] = (float)cp;
}

extern "C" void kernel_launch(void* const* d_in, const int* in_sizes, int n_in,
                              void* d_out, int out_size, void* d_ws, size_t ws_size,
                              hipStream_t stream) {
  (void)n_in; (void)out_size; (void)ws_size;
  const float* anchors = (const float*)d_in[0];
  const float* gtb     = (const float*)d_in[1];
  const int*   ishape  = (const int*)d_in[2];
  const int N = in_sizes[0] / 4;

  char* ws = (char*)d_ws;
  int* code = (int*)ws;                                       // N ints
  size_t off = ((size_t)N * 4 + 255) & ~(size_t)255;
  unsigned long long* gtBest = (unsigned long long*)(ws + off); off += GMAX * 8;
  int* cnts   = (int*)(ws + off); off += 64;                  // [posCnt, negCnt]
  int* posIdx = (int*)(ws + off); off += MAX_POS * 4;
  int* negIdx = (int*)(ws + off);

  init_ws<<<1, 64, 0, stream>>>(gtBest, cnts);

  long long nTiles = ((long long)N + 15) >> 4;
  int blocks = (int)((nTiles + WAVES - 1) / WAVES);
  if (blocks > 4096) blocks = 4096;
  iou_label_kernel<<<blocks, TPB, 0, stream>>>(anchors, gtb, ishape, N, code, gtBest);

  force_gt_kernel<<<1, 32, 0, stream>>>(gtBest, code);

  int b3 = (N + 255) / 256; if (b3 > 2048) b3 = 2048;
  sample_kernel<<<b3, 256, 0, stream>>>(code, N, cnts, posIdx, negIdx);

  finalize_kernel<<<1, 256, 0, stream>>>(anchors, gtb, code, posIdx, negIdx, cnts,
                                         (float*)d_out);
}